// S_I_29755533426927
// MI455X (gfx1250) — compile-verified
//
#include <hip/hip_runtime.h>

typedef unsigned int u32;
typedef u32 u32x4 __attribute__((ext_vector_type(4)));
typedef u32 u32x8 __attribute__((ext_vector_type(8)));

#define CHUNK 2048          // edges per chunk; 8 edges per thread (256 thr)
#define EDGE_BLOCKS 1280

// ---------------------------------------------------------------------------
// TDM: 1-D tensor DMA Global -> LDS (CDNA5 Tensor Data Mover).
// Builds a D# descriptor (groups 0 and 1, tensors up to 2D) in SGPRs and
// issues tensor_load_to_lds via inline asm (portable across ROCm7.2/clang-23
// builtin arity differences). Tracked with TENSORcnt.
//   group0: [1:0]=count=1, [63:32]=lds_addr, [120:64]=global_addr, [127:126]=type=2
//   group1: [17:16]=data_size(2 => 4B), [79:48]=tensor_dim0, [127:112]=tile_dim0,
//           [207:160]=tensor_dim0_stride
// tensor_dim0 = remaining elements so tail-chunk OOB reads return zero.
// ---------------------------------------------------------------------------
__device__ __forceinline__ void tdm_load_1d(u32 lds_off, const void* gptr,
                                            u32 tile_elems, u32 tensor_elems) {
  unsigned long long ga = (unsigned long long)(uintptr_t)gptr;
  u32x4 g0;
  g0.x = 1u;                               // count = 1 (valid descriptor)
  g0.y = lds_off;                          // LDS byte address of tile start
  g0.z = (u32)ga;                          // global_addr[31:0]
  g0.w = (u32)(ga >> 32) | (2u << 30);     // global_addr[56:32] | type=2

  u32x8 g1;
  g1.s0 = 2u << 16;                        // data_size = 4 bytes; no multicast/pad
  g1.s1 = tensor_elems << 16;              // tensor_dim0[15:0] at bit 48
  g1.s2 = tensor_elems >> 16;              // tensor_dim0[31:16]; tensor_dim1 = 0
  g1.s3 = tile_elems << 16;                // tile_dim0 at bit 112
  g1.s4 = 0u;                              // tile_dim1 = tile_dim2 = 0 (unused)
  g1.s5 = tensor_elems;                    // tensor_dim0_stride[31:0]
  g1.s6 = 0u;                              // tensor_dim0_stride[47:32]
  g1.s7 = 0u;                              // tensor_dim1_stride = 0

  asm volatile("tensor_load_to_lds %0, %1" :: "s"(g0), "s"(g1) : "memory");
}

// ---------------------------------------------------------------------------
// Kernel 1: per-node prep. s = (I==0)*susc ; iv = infect*I ; rowlogs = 0.
// Re-zeroing rowlogs every launch keeps kernel_launch deterministic.
// ---------------------------------------------------------------------------
__global__ void sir_prep_kernel(const float* __restrict__ I,
                                const float* __restrict__ sus,
                                const float* __restrict__ inf,
                                float* __restrict__ s,
                                float* __restrict__ iv,
                                float* __restrict__ rowlogs, int n) {
  int v = blockIdx.x * blockDim.x + threadIdx.x;
  if (v < n) {
    float Iv = I[v];
    s[v]       = (Iv == 0.0f) ? sus[v] : 0.0f;
    iv[v]      = inf[v] * Iv;
    rowlogs[v] = 0.0f;
  }
}

// ---------------------------------------------------------------------------
// Kernel 2: edge stage. TDM double-buffers 2048-edge chunks of src/dst
// indices into LDS; each thread consumes 8 contiguous edges via int4
// (ds_load_b128) reads, gathers s[src]*iv[dst] (L2-resident, 400KB each),
// and scatter-adds log(1-v) only when v != 0 (~9% of edges -> ~11x fewer
// atomics). Next chunk's DMA overlaps compute of the current chunk.
// ---------------------------------------------------------------------------
__global__ void sir_edge_kernel(const int* __restrict__ src,
                                const int* __restrict__ dst,
                                const float* __restrict__ s,
                                const float* __restrict__ iv,
                                float* __restrict__ rowlogs,
                                int nEdges, int nChunks) {
  __shared__ alignas(16) int buf[2][2 * CHUNK]; // [buffer][src | dst] = 32 KB
  const bool wave0 = (threadIdx.x < 32);        // one wave32 drives the TDM
  int c = blockIdx.x;
  if (c >= nChunks) return;

  // Prime buffer 0.
  if (wave0) {
    int base = c * CHUNK;
    tdm_load_1d((u32)(uintptr_t)&buf[0][0],     src + base, CHUNK, (u32)(nEdges - base));
    tdm_load_1d((u32)(uintptr_t)&buf[0][CHUNK], dst + base, CHUNK, (u32)(nEdges - base));
    __builtin_amdgcn_s_wait_tensorcnt(0);
  }
  __syncthreads();

  int cur = 0;
  for (;;) {
    int cn = c + gridDim.x;
    // Kick off next chunk's DMA into the other buffer (overlaps compute).
    if (cn < nChunks && wave0) {
      int nbase = cn * CHUNK;
      tdm_load_1d((u32)(uintptr_t)&buf[cur ^ 1][0],     src + nbase, CHUNK, (u32)(nEdges - nbase));
      tdm_load_1d((u32)(uintptr_t)&buf[cur ^ 1][CHUNK], dst + nbase, CHUNK, (u32)(nEdges - nbase));
    }

    // Consume current chunk: 8 contiguous edges per thread, int4 LDS reads.
    {
      const int4* sbuf = (const int4*)&buf[cur][0];
      const int4* dbuf = (const int4*)&buf[cur][CHUNK];
      int base = c * CHUNK;
      int tq   = threadIdx.x * 2;              // int4 slot; 2 per thread
#pragma unroll
      for (int q = 0; q < 2; ++q) {
        int4 sv = sbuf[tq + q];
        int4 dv = dbuf[tq + q];
        int  e0 = base + (tq + q) * 4;
        int  sidx[4] = {sv.x, sv.y, sv.z, sv.w};
        int  didx[4] = {dv.x, dv.y, dv.z, dv.w};
#pragma unroll
        for (int k = 0; k < 4; ++k) {
          float v = s[sidx[k]] * iv[didx[k]];
          // Tail chunk: TDM zero-fills OOB, but s[0]*iv[0] may be nonzero,
          // so the e<nEdges guard must gate the atomic.
          if ((e0 + k) < nEdges && v > 0.0f) {
            // log1p(-v) = log2(1-v) * ln2 ; v < 1 always (s,i in [0,1))
            atomicAdd(&rowlogs[sidx[k]],
                      __builtin_amdgcn_logf(1.0f - v) * 0.69314718055994531f);
          }
        }
      }
    }

    if (cn >= nChunks) break;
    if (wave0) __builtin_amdgcn_s_wait_tensorcnt(0);
    __syncthreads();
    c = cn;
    cur ^= 1;
  }
}

// ---------------------------------------------------------------------------
// Kernel 3: per-node finalize. p = 1-exp(rowlogs); deterministic hash draw
// (stand-in for jax.random key 42); I_new = max(I, dI).
// ---------------------------------------------------------------------------
__global__ void sir_node_kernel(const float* __restrict__ I,
                                const float* __restrict__ rowlogs,
                                float* __restrict__ out, int n) {
  int v = blockIdx.x * blockDim.x + threadIdx.x;
  if (v < n) {
    float r = rowlogs[v];
    // 1 - exp(r) = 1 - exp2(r * log2(e))
    float p = 1.0f - __builtin_amdgcn_exp2f(r * 1.4426950408889634f);
    u32 h = (u32)v * 0x9E3779B9u + 42u;
    h ^= h >> 16; h *= 0x85EBCA6Bu;
    h ^= h >> 13; h *= 0xC2B2AE35u;
    h ^= h >> 16;
    float u  = (float)(h >> 8) * (1.0f / 16777216.0f);
    float dI = (u < p) ? 1.0f : 0.0f;
    out[v] = fmaxf(I[v], dI);
  }
}

extern "C" void kernel_launch(void* const* d_in, const int* in_sizes, int n_in,
                              void* d_out, int out_size, void* d_ws, size_t ws_size,
                              hipStream_t stream) {
  const float* I   = (const float*)d_in[0];
  const float* sus = (const float*)d_in[1];
  const float* inf = (const float*)d_in[2];
  const int*   src = (const int*)d_in[3];
  const int*   dst = (const int*)d_in[4];
  int n = in_sizes[0];
  int E = in_sizes[3];

  float* s  = (float*)d_ws;    // n floats
  float* iv = s + n;           // n floats
  float* rl = iv + n;          // n floats (accumulator)
  float* out = (float*)d_out;

  int nodeBlocks = (n + 255) / 256;
  sir_prep_kernel<<<nodeBlocks, 256, 0, stream>>>(I, sus, inf, s, iv, rl, n);

  int nChunks = (E + CHUNK - 1) / CHUNK;
  int grid = nChunks < EDGE_BLOCKS ? nChunks : EDGE_BLOCKS;
  sir_edge_kernel<<<grid, 256, 0, stream>>>(src, dst, s, iv, rl, E, nChunks);

  sir_node_kernel<<<nodeBlocks, 256, 0, stream>>>(I, rl, out, n);
}